// medror_34428457845315
// MI455X (gfx1250) — compile-verified
//
#include <hip/hip_runtime.h>
#include <cstdint>

// Problem constants (fixed by the reference)
constexpr int kH  = 64;
constexpr int kW  = 2048;
constexpr int kC  = 8;          // 4 * N_ECHOES
constexpr int kHW = kH * kW;
constexpr int kSEARCH = 9;
constexpr int kPAD    = 4;      // (SEARCH-1)/2
constexpr int kKNN    = 9;
constexpr float kTS   = 3.0f * 0.008f;   // 0.024

// Tiling: one 256-thread block (8 wave32) handles a 32(w) x 8(h) pixel tile.
constexpr int TW = 32;
constexpr int TH = 8;
constexpr int LW = TW + 2 * kPAD;   // 40 floats = 160 B per row (16B aligned)
constexpr int LH = TH + 2 * kPAD;   // 16

__global__ __launch_bounds__(256) void medror_knn_kernel(const float* __restrict__ x,
                                                         float* __restrict__ out) {
    // SoA layout: [channel][row][col] -> DS loads are lane-stride-4B (conflict free)
    __shared__ float q[3][LH][LW];

    const int tx  = threadIdx.x;          // 0..31 (one wave per tile row)
    const int ty  = threadIdx.y;          // 0..7
    const int tid = ty * TW + tx;
    const int b   = blockIdx.z;

    const int h0 = blockIdx.y * TH - kPAD;   // top-left of LDS window in image coords
    const int w0 = blockIdx.x * TW - kPAD;

    // First-echo point channels start at channel 2 (N_ECHOES = 2)
    const float* baseq = x + (size_t)b * kC * kHW + (size_t)2 * kHW;
    const unsigned long long sbase = (unsigned long long)(uintptr_t)baseq;

    // ---- Stage window into LDS with the CDNA5 async global->LDS DMA path ----
    const bool interior = (h0 >= 0) & (h0 + LH <= kH) & (w0 >= 0) & (w0 + LW <= kW);
    if (interior) {
        // Fast path: whole window in-bounds; move 16 B per lane.
        // 3ch x 16 rows x 10 xmits = 480 b128 transfers; both addresses 16B aligned.
        constexpr int NCHUNK = 3 * LH * (LW / 4);   // 480
        for (int idx = tid; idx < NCHUNK; idx += 256) {
            const int c   = idx / (LH * (LW / 4));
            const int rem = idx - c * (LH * (LW / 4));
            const int r   = rem / (LW / 4);
            const int ch  = rem - r * (LW / 4);
            const uint32_t loff = (uint32_t)(uintptr_t)&q[c][r][ch * 4];
            const uint32_t goff = (uint32_t)((c * kHW + (h0 + r) * kW + w0 + ch * 4) * 4);
            asm volatile("global_load_async_to_lds_b128 %0, %1, %2"
                         :: "v"(loff), "v"(goff), "s"(sbase)
                         : "memory");
        }
    } else {
        // Border path: per-dword async load when in bounds, ds_store zero otherwise
        // (implements jnp.pad zero padding).
        constexpr int TOT = 3 * LH * LW;   // 1920 dwords
        for (int idx = tid; idx < TOT; idx += 256) {
            const int c   = idx / (LH * LW);
            const int rem = idx - c * (LH * LW);
            const int r   = rem / LW;
            const int col = rem - r * LW;
            const int gh  = h0 + r;
            const int gw  = w0 + col;
            if ((unsigned)gh < (unsigned)kH && (unsigned)gw < (unsigned)kW) {
                const uint32_t loff = (uint32_t)(uintptr_t)&q[c][r][col];
                const uint32_t goff = (uint32_t)((c * kHW + gh * kW + gw) * 4);
                asm volatile("global_load_async_to_lds_b32 %0, %1, %2"
                             :: "v"(loff), "v"(goff), "s"(sbase)
                             : "memory");
            } else {
                q[c][r][col] = 0.0f;
            }
        }
    }

    // ---- Per-pixel operands (independent of LDS): overlap with the async DMA ----
    const int gw  = blockIdx.x * TW + tx;
    const int gh  = blockIdx.y * TH + ty;
    const int pix = gh * kW + gw;
    const float* bx = x + (size_t)b * kC * kHW;

    const float r0  = bx[0 * kHW + pix];
    const float r1  = bx[1 * kHW + pix];
    const float p0x = bx[2 * kHW + pix];
    const float p0y = bx[3 * kHW + pix];
    const float p0z = bx[4 * kHW + pix];
    const float p1x = bx[5 * kHW + pix];
    const float p1y = bx[6 * kHW + pix];
    const float p1z = bx[7 * kHW + pix];

    const float t0 = r0 * kTS, t02 = t0 * t0;
    const float t1 = r1 * kTS, t12 = t1 * t1;

    // Drain this wave's async transfers, then make them visible workgroup-wide.
    asm volatile("s_wait_asynccnt 0" ::: "memory");
    __syncthreads();

    // Z = exact-zero distances, C = distances in (0, thresh].
    // 9-NN survivor count == min(C, max(0, 9 - Z))  (no top-k needed).
    int z0 = 0, c0 = 0, z1 = 0, c1 = 0;

    for (int di = 0; di < kSEARCH; ++di) {
        const float* row0 = &q[0][ty + di][tx];
        const float* row1 = &q[1][ty + di][tx];
        const float* row2 = &q[2][ty + di][tx];
#pragma unroll
        for (int dj = 0; dj < kSEARCH; ++dj) {
            const float qx = row0[dj];
            const float qy = row1[dj];
            const float qz = row2[dj];

            float dx = p0x - qx, dy = p0y - qy, dz = p0z - qz;
            float d2 = fmaf(dz, dz, fmaf(dy, dy, dx * dx));
            if (d2 == 0.0f)      z0++;
            else if (d2 <= t02)  c0++;

            dx = p1x - qx; dy = p1y - qy; dz = p1z - qz;
            d2 = fmaf(dz, dz, fmaf(dy, dy, dx * dx));
            if (d2 == 0.0f)      z1++;
            else if (d2 <= t12)  c1++;
        }
    }

    int slots0 = kKNN - z0; if (slots0 < 0) slots0 = 0;
    int slots1 = kKNN - z1; if (slots1 < 0) slots1 = 0;
    const int cnt0 = c0 < slots0 ? c0 : slots0;
    const int cnt1 = c1 < slots1 ? c1 : slots1;

    float* ob = out + (size_t)b * 2 * kHW;
    ob[pix]       = (cnt0 < 3) ? 1000.0f : -1000.0f;
    ob[kHW + pix] = (cnt1 < 3) ? 1000.0f : -1000.0f;
}

extern "C" void kernel_launch(void* const* d_in, const int* in_sizes, int n_in,
                              void* d_out, int out_size, void* d_ws, size_t ws_size,
                              hipStream_t stream) {
    (void)n_in; (void)out_size; (void)d_ws; (void)ws_size;
    const float* x = (const float*)d_in[0];
    float* out = (float*)d_out;

    const int B = in_sizes[0] / (kC * kHW);   // = 2 for the reference shapes
    dim3 grid(kW / TW, kH / TH, B);
    dim3 block(TW, TH, 1);
    medror_knn_kernel<<<grid, block, 0, stream>>>(x, out);
}